// OmegaRule_41635412968170
// MI455X (gfx1250) — compile-verified
//
#include <hip/hip_runtime.h>
#include <math.h>

// ---------------- problem constants ----------------
#define BQ      8
#define SQ      512
#define DK      64
#define DV      128
#define CTX     64
#define PHI     2145               // 1 + 64 + 2080
#define PHI_PAD 2160               // padded to /16 and /4
#define SPAD    (SQ + CTX - 1)     // 575 rows per batch (left-padded)
#define MOMC    0.95f
#define EPSC    1e-8f
#define CKB     48                 // prederr A-panel chunk (floats per row)

typedef __attribute__((ext_vector_type(2))) float v2f;
typedef __attribute__((ext_vector_type(8))) float v8f;
typedef __attribute__((ext_vector_type(4))) int   v4i;

// D = A(16x4) * B(4x16) + C, fp32 WMMA on CDNA5
__device__ __forceinline__ v8f wmma_f32x4(v2f a, v2f b, v8f c) {
    return __builtin_amdgcn_wmma_f32_16x16x4_f32(
        /*neg_a=*/false, a, /*neg_b=*/false, b,
        /*c_mod=*/(short)0, c, /*reuse_a=*/false, /*reuse_b=*/false);
}

__device__ __forceinline__ float sigmoidf_(float x) { return 1.0f / (1.0f + expf(-x)); }
__device__ __forceinline__ float siluf_(float x)    { return x * sigmoidf_(x); }

// -------- async global -> LDS staging (CDNA5 ASYNCcnt path) --------
#if defined(__has_builtin)
#  if __has_builtin(__builtin_amdgcn_global_load_async_to_lds_b128)
#    define OMEGA_HAVE_ASYNC 1
#  endif
#endif

// builtin signature (from hipcc diagnostic): (v4i addrspace(1)*, v4i addrspace(3)*, imm, imm)
typedef __attribute__((address_space(1))) v4i g_v4i;
typedef __attribute__((address_space(3))) v4i l_v4i;

__device__ __forceinline__ void async_ld_b128(const float* g, float* l) {
#ifdef OMEGA_HAVE_ASYNC
    __builtin_amdgcn_global_load_async_to_lds_b128((g_v4i*)g, (l_v4i*)l, 0, 0);
#else
    *(float4*)l = *(const float4*)g;   // synchronous fallback
#endif
}

__device__ __forceinline__ void panel_wait_sync() {
#ifdef OMEGA_HAVE_ASYNC
#  if defined(__has_builtin) && __has_builtin(__builtin_amdgcn_s_wait_asynccnt)
    __builtin_amdgcn_s_wait_asynccnt(0);
#  else
    asm volatile("s_wait_asynccnt 0x0" ::: "memory");
#  endif
#endif
    __syncthreads();
}

// ================= prep 1: degree-2 poly features (padded) =================
__global__ void omega_prep_feats(const float* __restrict__ keys,
                                 const float* __restrict__ queries,
                                 float* __restrict__ kp_pad,
                                 float* __restrict__ qp) {
    __shared__ float xk[DK];
    __shared__ float xq[DK];
    const int row = blockIdx.x;          // 0 .. B*SPAD-1
    const int b   = row / SPAD;
    const int rr  = row % SPAD;
    const int src = rr - (CTX - 1);
    if (threadIdx.x < DK) {
        xk[threadIdx.x] = (src >= 0) ? keys[((size_t)(b * SQ + src)) * DK + threadIdx.x] : 0.0f;
        xq[threadIdx.x] = (src >= 0) ? queries[((size_t)(b * SQ + src)) * DK + threadIdx.x] : 0.0f;
    }
    __syncthreads();
    float* kout = kp_pad + (size_t)row * PHI_PAD;
    for (int f = threadIdx.x; f < PHI_PAD; f += blockDim.x) {
        float vk = 0.0f, vq = 0.0f;
        if (src >= 0 && f < PHI) {
            if (f == 0)          { vk = 1.0f; vq = 1.0f; }
            else if (f < 1 + DK) { vk = xk[f - 1]; vq = xq[f - 1]; }
            else {
                int q = f - (1 + DK);
                int i = 0, rem = q;
                while (rem >= DK - i) { rem -= DK - i; ++i; }   // triu row-major
                int j = i + rem;
                vk = xk[i] * xk[j];
                vq = xq[i] * xq[j];
            }
        }
        kout[f] = vk;
        if (src >= 0) qp[((size_t)(b * SQ + src)) * PHI_PAD + f] = vq;
    }
}

// ================= prep 2: state init (vpad, M, buf, nrm) =================
__global__ void omega_prep_state(const float* __restrict__ values,
                                 const float* __restrict__ M0,
                                 float* __restrict__ vpad,
                                 float* __restrict__ Mw,
                                 float* __restrict__ buf,
                                 float* __restrict__ nrm) {
    const int NV = BQ * SPAD * DV;          // 588800
    const int NM = DV * PHI_PAD;            // 276480
    const int TOT = NV + 2 * NM + SQ;
    for (int idx = blockIdx.x * blockDim.x + threadIdx.x; idx < TOT;
         idx += gridDim.x * blockDim.x) {
        if (idx < NV) {
            int v = idx & (DV - 1);
            int r = idx >> 7;
            int b = r / SPAD, rr = r % SPAD;
            int src = rr - (CTX - 1);
            vpad[idx] = (src >= 0) ? values[((size_t)(b * SQ + src)) * DV + v] : 0.0f;
        } else if (idx < NV + NM) {
            int k = idx - NV;
            int col = k % PHI_PAD, rowv = k / PHI_PAD;
            Mw[k] = (col < PHI) ? M0[(size_t)rowv * PHI + col] : 0.0f;
        } else if (idx < NV + 2 * NM) {
            buf[idx - NV - NM] = 0.0f;
        } else {
            nrm[idx - NV - 2 * NM] = 0.0f;
        }
    }
}

// ================= prep 3: gates gamma[t][b*64+c], alpha[t], lr[t] =========
__global__ void omega_prep_gates(const float* __restrict__ keys,
                                 const float* __restrict__ w_decay,
                                 const float* __restrict__ b_decay,
                                 const float* __restrict__ w_alpha,
                                 const float* __restrict__ b_alpha,
                                 float* __restrict__ gamma,
                                 float* __restrict__ alpha,
                                 float* __restrict__ lr) {
    __shared__ float wsh[BQ * CTX];
    __shared__ float ssum[BQ];
    __shared__ float lsh[BQ];
    const int t = blockIdx.x;
    const int i = threadIdx.x;         // 0..511
    const int b = i >> 6, c = i & 63;
    const int src = t + c - (CTX - 1);
    float w = 0.0f;
    if (src >= 0) {
        const float* kr = keys + ((size_t)(b * SQ + src)) * DK;
        float d = b_decay[0];
        #pragma unroll 8
        for (int k = 0; k < DK; ++k) d += kr[k] * w_decay[k];
        w = sigmoidf_(d);
    }
    wsh[i] = w;
    __syncthreads();
    if (c == 0) {
        float s = 0.0f;
        for (int j = 0; j < CTX; ++j) s += wsh[b * CTX + j];
        ssum[b] = s;
        const float* kr = keys + ((size_t)(b * SQ + t)) * DK;
        float d = b_alpha[0];
        for (int k = 0; k < DK; ++k) d += kr[k] * w_alpha[k];
        lsh[b] = d;
    }
    __syncthreads();
    gamma[(size_t)t * (BQ * CTX) + i] = w / (ssum[b] + 1e-8f);
    if (i == 0) {
        float m = 0.0f;
        for (int j = 0; j < BQ; ++j) m += lsh[j];
        float a = sigmoidf_(m * (1.0f / BQ));
        alpha[t] = a;
        lr[t]    = 0.1f * (1.0f - a);
    }
}

// ===== step A: ET[v][r] = gamma[t][r]*(silu(kp_w @ M^T) - v_w)/B ==========
// 1 block = 1 row-tile (16 kp rows shared by all 8 waves) x 8 dv-tiles.
// A-panel double-buffered in LDS via async global->LDS b128.
__global__ __launch_bounds__(256)
void omega_step_prederr(const float* __restrict__ kp_pad,
                        const float* __restrict__ vpad,
                        const float* __restrict__ Mw,
                        const float* __restrict__ gamma,
                        float* __restrict__ ET, int t) {
    __shared__ float apanel[2][16 * CKB];    // 2 x 3 KB
    const int tid  = threadIdx.x;
    const int wave = tid >> 5;
    const int lane = tid & 31;
    const int rt = blockIdx.x;               // 0..31
    const int ct = wave;                     // 0..7
    const int r0 = rt * 16;
    const int m  = lane & 15;
    const int kh = lane >> 4;

    // async fill mapping: 192 x b128 per chunk (16 rows x 12 quads)
    const int irow = tid / 12;
    const int iq   = tid - irow * 12;
    const float* gsrc = kp_pad;              // valid dummy
    if (tid < 192) {
        int r = r0 + irow;
        int b = r >> 6, c = r & 63;
        gsrc = kp_pad + ((size_t)(b * SPAD + t + c)) * PHI_PAD + iq * 4;
    }
    const float* brow = Mw + ((size_t)(ct * 16 + m)) * PHI_PAD + 2 * kh;

    v8f acc = {};
    if (tid < 192) async_ld_b128(gsrc, &apanel[0][irow * CKB + iq * 4]);
    panel_wait_sync();
    int buf = 0;
    const int NCH = PHI_PAD / CKB;           // 45
    for (int ch = 0; ch < NCH; ++ch) {
        if (ch + 1 < NCH && tid < 192)
            async_ld_b128(gsrc + (size_t)(ch + 1) * CKB,
                          &apanel[buf ^ 1][irow * CKB + iq * 4]);
        const float* ap = &apanel[buf][m * CKB + 2 * kh];
        const float* bp = brow + (size_t)ch * CKB;
        #pragma unroll
        for (int kkl = 0; kkl < CKB; kkl += 4) {
            v2f a  = *(const v2f*)(ap + kkl);
            v2f bb = *(const v2f*)(bp + kkl);
            acc = wmma_f32x4(a, bb, acc);
        }
        panel_wait_sync();
        buf ^= 1;
    }
    const float* gt = gamma + (size_t)t * (BQ * CTX);
    #pragma unroll
    for (int j = 0; j < 8; ++j) {
        int rr = r0 + j + 8 * kh;
        int bb_ = rr >> 6, cc = rr & 63;
        float pred = siluf_(acc[j]);
        float vv = vpad[((size_t)(bb_ * SPAD + t + cc)) * DV + ct * 16 + m];
        // store TRANSPOSED for the grad GEMM's A operand
        ET[(size_t)(ct * 16 + m) * SQ + rr] = (pred - vv) * gt[rr] * (1.0f / BQ);
    }
}

// ===== step B: grad = ET @ kp_w ; fused Muon buffer + norm partials =======
// 1 block = 1 phi-tile (16 kp cols, 512 rows staged once in 32KB LDS) x 8 dv-tiles
__global__ __launch_bounds__(256)
void omega_step_grad(const float* __restrict__ kp_pad,
                     const float* __restrict__ ET,
                     float* __restrict__ buf,
                     float* __restrict__ gbuf,
                     float* __restrict__ nrm, int t) {
    __shared__ float Bp[512 * 16];           // 32 KB
    const int tid  = threadIdx.x;
    const int wave = tid >> 5;
    const int lane = tid & 31;
    const int pt = blockIdx.x;               // 0..134
    const int vt = wave;                     // 0..7
    const int v0 = vt * 16, p0 = pt * 16;
    const int m  = lane & 15;
    const int kh = lane >> 4;

    // stage B-panel: kp cols [p0,p0+16) for all 512 window rows (async b128)
    for (int idx = tid; idx < 2048; idx += 256) {
        int row = idx >> 2, q = idx & 3;
        int b = row >> 6, c = row & 63;
        const float* g = kp_pad + ((size_t)(b * SPAD + t + c)) * PHI_PAD + p0 + q * 4;
        async_ld_b128(g, &Bp[row * 16 + q * 4]);
    }
    panel_wait_sync();

    const float* arow = ET + (size_t)(v0 + m) * SQ + 2 * kh;
    v8f acc = {};
    #pragma unroll 8
    for (int kk = 0; kk < BQ * CTX; kk += 4) {
        int k0 = kk + 2 * kh;
        v2f a = *(const v2f*)(arow + kk);
        v2f bb;
        bb.x = Bp[k0 * 16 + m];
        bb.y = Bp[(k0 + 1) * 16 + m];
        acc = wmma_f32x4(a, bb, acc);
    }
    float local = 0.0f;
    #pragma unroll
    for (int j = 0; j < 8; ++j) {
        int v = v0 + j + 8 * kh;
        size_t idx = (size_t)v * PHI_PAD + p0 + m;
        float grad = acc[j];
        float bnew = MOMC * buf[idx] + grad;       // Muon momentum
        float g    = grad + MOMC * bnew;           // Nesterov
        buf[idx]  = bnew;
        gbuf[idx] = g;
        local += g * g;
    }
    #pragma unroll
    for (int off = 16; off > 0; off >>= 1) local += __shfl_xor(local, off, 32);
    if (lane == 0) atomicAdd(&nrm[t], local);
}

// ===== step C: M = (1-alpha)*M - lr * g / sqrt(||g||_F + eps) =============
__global__ void omega_step_update(float* __restrict__ Mw,
                                  const float* __restrict__ gbuf,
                                  const float* __restrict__ alpha,
                                  const float* __restrict__ lr,
                                  const float* __restrict__ nrm, int t) {
    const float a   = alpha[t];
    const float inv = lr[t] / sqrtf(sqrtf(nrm[t]) + EPSC);
    const int idx = blockIdx.x * blockDim.x + threadIdx.x;
    if (idx < DV * PHI_PAD)
        Mw[idx] = (1.0f - a) * Mw[idx] - inv * gbuf[idx];
}

// ===== step D: out[:, t, :] = silu(qp_t @ M_new^T), 1 block, 8 waves ======
__global__ __launch_bounds__(256)
void omega_step_out(const float* __restrict__ qp,
                    const float* __restrict__ Mw,
                    float* __restrict__ out, int t) {
    const int ct   = threadIdx.x >> 5;          // dv tile 0..7
    const int lane = threadIdx.x & 31;
    const int m  = lane & 15;
    const int kh = lane >> 4;
    const bool rowValid = (m < BQ);
    const float* arow = rowValid ? (qp + ((size_t)(m * SQ + t)) * PHI_PAD + 2 * kh) : qp;
    const float* brow = Mw + ((size_t)(ct * 16 + m)) * PHI_PAD + 2 * kh;
    v8f acc = {};
    for (int kk = 0; kk < PHI_PAD; kk += 4) {
        v2f a = rowValid ? *(const v2f*)(arow + kk) : (v2f){0.0f, 0.0f};
        v2f bb = *(const v2f*)(brow + kk);
        acc = wmma_f32x4(a, bb, acc);
    }
    if (kh == 0) {                               // rows 0..7 == batches
        #pragma unroll
        for (int j = 0; j < 8; ++j) {
            out[((size_t)(j * SQ + t)) * DV + ct * 16 + m] = siluf_(acc[j]);
        }
    }
}

// =========================== launcher =====================================
extern "C" void kernel_launch(void* const* d_in, const int* in_sizes, int n_in,
                              void* d_out, int out_size, void* d_ws, size_t ws_size,
                              hipStream_t stream) {
    const float* keys    = (const float*)d_in[0];
    const float* values  = (const float*)d_in[1];
    const float* queries = (const float*)d_in[2];
    const float* M0      = (const float*)d_in[3];
    const float* w_decay = (const float*)d_in[4];
    const float* b_decay = (const float*)d_in[5];
    const float* w_alpha = (const float*)d_in[6];
    const float* b_alpha = (const float*)d_in[7];
    float* out = (float*)d_out;

    const size_t N_KP  = (size_t)BQ * SPAD * PHI_PAD;   // 9,936,000
    const size_t N_QP  = (size_t)BQ * SQ * PHI_PAD;     // 8,847,360
    const size_t N_VP  = (size_t)BQ * SPAD * DV;        //   588,800
    const size_t N_M   = (size_t)DV * PHI_PAD;          //   276,480
    const size_t N_E   = (size_t)DV * SQ;               //    65,536
    const size_t N_G   = (size_t)SQ * BQ * CTX;         //   262,144
    float* kp_pad = (float*)d_ws;
    float* qp     = kp_pad + N_KP;
    float* vpad   = qp + N_QP;
    float* Mw     = vpad + N_VP;
    float* buf    = Mw + N_M;
    float* gbuf   = buf + N_M;
    float* ET     = gbuf + N_M;
    float* gamma  = ET + N_E;
    float* alpha  = gamma + N_G;
    float* lrv    = alpha + SQ;
    float* nrm    = lrv + SQ;
    const size_t needBytes = ((size_t)(nrm + SQ - kp_pad)) * sizeof(float);
    if (ws_size < needBytes) return;

    omega_prep_feats<<<BQ * SPAD, 256, 0, stream>>>(keys, queries, kp_pad, qp);
    {
        const int TOT = BQ * SPAD * DV + 2 * DV * PHI_PAD + SQ;
        omega_prep_state<<<(TOT + 255) / 256, 256, 0, stream>>>(values, M0, vpad, Mw, buf, nrm);
    }
    omega_prep_gates<<<SQ, BQ * CTX, 0, stream>>>(keys, w_decay, b_decay,
                                                  w_alpha, b_alpha, gamma, alpha, lrv);

    for (int t = 0; t < SQ; ++t) {
        omega_step_prederr<<<32, 256, 0, stream>>>(kp_pad, vpad, Mw, gamma, ET, t);
        omega_step_grad<<<135, 256, 0, stream>>>(kp_pad, ET, buf, gbuf, nrm, t);
        omega_step_update<<<(DV * PHI_PAD + 255) / 256, 256, 0, stream>>>(Mw, gbuf, alpha, lrv, nrm, t);
        omega_step_out<<<1, 256, 0, stream>>>(qp, Mw, out, t);
    }
}